// GCN_16552803959294
// MI455X (gfx1250) — compile-verified
//
#include <hip/hip_runtime.h>

#define Nn 100000
#define Ee 1000000
#define INF 128
#define HID 128
#define CLS 64

typedef __attribute__((ext_vector_type(2))) float v2f;
typedef __attribute__((ext_vector_type(8))) float v8f;

__device__ __forceinline__ void atomic_add_f32(float* p, float v) {
    __hip_atomic_fetch_add(p, v, __ATOMIC_RELAXED, __HIP_MEMORY_SCOPE_AGENT);
}

// ---------------- degree / norm ----------------
__global__ void k_zero_deg(int* __restrict__ deg) {
    int i = blockIdx.x * blockDim.x + threadIdx.x;
    if (i < Nn) deg[i] = 0;
}

__global__ void k_count_deg(const int* __restrict__ dst, int* __restrict__ deg) {
    int e = blockIdx.x * blockDim.x + threadIdx.x;
    if (e < Ee) atomicAdd(&deg[dst[e]], 1);
}

__global__ void k_dinv(const int* __restrict__ deg, float* __restrict__ dinv) {
    int i = blockIdx.x * blockDim.x + threadIdx.x;
    if (i < Nn) {
        float d = (float)deg[i] + 1.0f;   // +1 self-loop; always >= 1
        dinv[i] = 1.0f / sqrtf(d);
    }
}

// ---------------- GEMM: H = act(X) @ W  (X: [N,128], W: [128,DOUT]) ----------------
// block = (DOUT/16) waves, each wave computes a 16x16 tile of 16 rows.
template <int DOUT, bool RELU_IN>
__global__ void k_gemm_wmma(const float* __restrict__ X,
                            const float* __restrict__ W,
                            float* __restrict__ H) {
    __shared__ float lds_x[16 * 132];      // 16 rows of K=128, padded stride 132
    __shared__ float lds_w[128 * DOUT];

    const int tid     = threadIdx.x;
    const int lane    = tid & 31;
    const int wave    = tid >> 5;
    const int rowBase = blockIdx.x * 16;
    const int colBase = wave * 16;
    const int nthr    = (DOUT / 16) * 32;

    // stage X tile (apply ReLU of previous layer here)
    for (int i = tid; i < 16 * 128; i += nthr) {
        int r = i >> 7, c = i & 127;
        float v = X[(size_t)(rowBase + r) * 128 + c];
        if (RELU_IN) v = fmaxf(v, 0.0f);
        lds_x[r * 132 + c] = v;
    }
    // stage full W
    for (int i = tid; i < 128 * DOUT; i += nthr) lds_w[i] = W[i];
    __syncthreads();

    const int r    = lane & 15;      // row within tile (A) / col within tile (B,D)
    const int half = lane >> 4;
    const int koff = half * 2;

    v8f acc = {};
#pragma unroll
    for (int kk = 0; kk < 128; kk += 4) {
        v2f a = *(const v2f*)(lds_x + r * 132 + kk + koff);           // A(r, kk+koff..+1)
        v2f b;
        b.x = lds_w[(kk + koff) * DOUT + colBase + r];                 // B(kk+koff,   n=r)
        b.y = lds_w[(kk + koff + 1) * DOUT + colBase + r];             // B(kk+koff+1, n=r)
        acc = __builtin_amdgcn_wmma_f32_16x16x4_f32(
            /*neg_a=*/false, a, /*neg_b=*/false, b,
            /*c_mod=*/(short)0, acc, /*reuse_a=*/false, /*reuse_b=*/false);
    }

#pragma unroll
    for (int v = 0; v < 8; ++v) {
        H[(size_t)(rowBase + v + half * 8) * DOUT + colBase + r] = acc[v];
    }
}

// ---------------- aggregation init: out[i] = h[i]*dinv[i]^2 + b  (self-loop + bias) ----
template <int DOUT>
__global__ void k_init_agg(const float* __restrict__ h,
                           const float* __restrict__ b,
                           const float* __restrict__ dinv,
                           float* __restrict__ out) {
    const int per = DOUT / 4;
    int t = blockIdx.x * blockDim.x + threadIdx.x;
    if (t >= Nn * per) return;
    int i  = t / per;
    int c4 = t - i * per;
    float w = dinv[i];
    w = w * w;
    float4 hv = ((const float4*)h)[t];
    float4 bb = ((const float4*)b)[c4];
    float4 o;
    o.x = hv.x * w + bb.x;
    o.y = hv.y * w + bb.y;
    o.z = hv.z * w + bb.z;
    o.w = hv.w * w + bb.w;
    ((float4*)out)[t] = o;
}

// ---------------- edge scatter: out[dst] += h[src] * dinv[src]*dinv[dst] --------------
template <int DOUT>
__global__ void k_scatter(const float* __restrict__ h,
                          const int* __restrict__ src,
                          const int* __restrict__ dst,
                          const float* __restrict__ dinv,
                          float* __restrict__ out) {
    const int per = DOUT / 4;   // threads per edge (power of 2)
    long long t = (long long)blockIdx.x * blockDim.x + threadIdx.x;
    if (t >= (long long)Ee * per) return;
    int e = (int)(t / per);
    int c = (int)(t - (long long)e * per);
    int s = src[e];
    int d = dst[e];
    float w = dinv[s] * dinv[d];
    float4 hv = ((const float4*)h)[(size_t)s * per + c];
    float* o = out + (size_t)d * DOUT + c * 4;
    atomic_add_f32(o + 0, hv.x * w);
    atomic_add_f32(o + 1, hv.y * w);
    atomic_add_f32(o + 2, hv.z * w);
    atomic_add_f32(o + 3, hv.w * w);
}

extern "C" void kernel_launch(void* const* d_in, const int* in_sizes, int n_in,
                              void* d_out, int out_size, void* d_ws, size_t ws_size,
                              hipStream_t stream) {
    const float* X   = (const float*)d_in[0];
    const float* W1  = (const float*)d_in[1];
    const float* b1  = (const float*)d_in[2];
    const float* W2  = (const float*)d_in[3];
    const float* b2  = (const float*)d_in[4];
    const float* W3  = (const float*)d_in[5];
    const float* b3  = (const float*)d_in[6];
    const int*   ei  = (const int*)d_in[7];   // [2, E]
    const int*   src = ei;
    const int*   dst = ei + Ee;
    float* out = (float*)d_out;

    float* bufA = (float*)d_ws;                       // N x 128
    float* bufB = bufA + (size_t)Nn * 128;            // N x 128
    int*   deg  = (int*)(bufB + (size_t)Nn * 128);    // N
    float* dinv = (float*)(deg + Nn);                 // N

    const int T = 256;
    // degree + normalization
    k_zero_deg<<<(Nn + T - 1) / T, T, 0, stream>>>(deg);
    k_count_deg<<<(Ee + T - 1) / T, T, 0, stream>>>(dst, deg);
    k_dinv<<<(Nn + T - 1) / T, T, 0, stream>>>(deg, dinv);

    const int gGemm = Nn / 16;        // 6250, exact

    // ---- layer 1 ----
    k_gemm_wmma<128, false><<<gGemm, 256, 0, stream>>>(X, W1, bufA);
    k_init_agg<128><<<(Nn * 32 + T - 1) / T, T, 0, stream>>>(bufA, b1, dinv, bufB);
    k_scatter<128><<<(int)(((long long)Ee * 32 + T - 1) / T), T, 0, stream>>>(bufA, src, dst, dinv, bufB);

    // ---- layer 2 (relu fused into A-tile load) ----
    k_gemm_wmma<128, true><<<gGemm, 256, 0, stream>>>(bufB, W2, bufA);
    k_init_agg<128><<<(Nn * 32 + T - 1) / T, T, 0, stream>>>(bufA, b2, dinv, bufB);
    k_scatter<128><<<(int)(((long long)Ee * 32 + T - 1) / T), T, 0, stream>>>(bufA, src, dst, dinv, bufB);

    // ---- layer 3 -> d_out ----
    k_gemm_wmma<64, true><<<gGemm, 128, 0, stream>>>(bufB, W3, bufA);
    k_init_agg<64><<<(Nn * 16 + T - 1) / T, T, 0, stream>>>(bufA, b3, dinv, out);
    k_scatter<64><<<(int)(((long long)Ee * 16 + T - 1) / T), T, 0, stream>>>(bufA, src, dst, dinv, out);
}